// SAGEConv_61735859912840
// MI455X (gfx1250) — compile-verified
//
#include <hip/hip_runtime.h>

typedef __attribute__((ext_vector_type(2))) float v2f;
typedef __attribute__((ext_vector_type(8))) float v8f;

// ---------------------------------------------------------------------------
// Constants (match reference)
// ---------------------------------------------------------------------------
constexpr int IN_FEAT  = 128;
constexpr int KDIM     = 2 * IN_FEAT;   // 256 (concat width)
constexpr int OUT_FEAT = 128;
constexpr int TILE_M   = 16;
constexpr int LDF      = KDIM + 4;      // padded LDS stride (bank-conflict avoidance)

// Relaxed agent-scope f32 atomic add -> global_atomic_add_f32 on gfx1250
__device__ __forceinline__ void atomic_add_f32(float* p, float v) {
    __hip_atomic_fetch_add(p, v, __ATOMIC_RELAXED, __HIP_MEMORY_SCOPE_AGENT);
}

// ---------------------------------------------------------------------------
// Kernel 1: edge-parallel gather + segment-sum (+ degree count)
// One wave32 per edge: 32 lanes x 4 iterations cover 128 features, coalesced.
// ---------------------------------------------------------------------------
__global__ __launch_bounds__(256) void sage_scatter(
    const float* __restrict__ h_s, const int* __restrict__ src,
    const int* __restrict__ dst, float* __restrict__ nsum,
    float* __restrict__ deg, int n_edges)
{
    const int waves_total = (gridDim.x * blockDim.x) >> 5;
    const int wave = (blockIdx.x * blockDim.x + threadIdx.x) >> 5;
    const int lane = threadIdx.x & 31;

    for (int e = wave; e < n_edges; e += waves_total) {
        const int s = src[e];
        const int d = dst[e];
        const float* hs = h_s + (size_t)s * IN_FEAT;
        float*       ns = nsum + (size_t)d * IN_FEAT;
#pragma unroll
        for (int i = 0; i < IN_FEAT / 32; ++i) {
            atomic_add_f32(&ns[lane + i * 32], hs[lane + i * 32]);
        }
        if (lane == 0) atomic_add_f32(&deg[d], 1.0f);
    }
}

// ---------------------------------------------------------------------------
// Kernel 2: fused mean + concat + Linear via V_WMMA_F32_16X16X4_F32.
// Block: 256 threads = 8 waves; each wave computes a 16x16 tile of out.
// feat tile [16 rows x 256 cols] staged in padded LDS.
//
// WMMA f32 16x16x4 operand layouts (ISA 7.12.2):
//   A (16x4):  lane L: row M = L&15; VGPR0/1 hold K = 2*(L>>4) + {0,1}
//   B (4x16):  lane L: col N = L&15; VGPR0/1 hold K = 2*(L>>4) + {0,1}
//   C/D:       lane L: col N = L&15; VGPR v holds row M = v + 8*(L>>4)
// ---------------------------------------------------------------------------
__global__ __launch_bounds__(256) void sage_gemm(
    const float* __restrict__ h_d, const float* __restrict__ nsum,
    const float* __restrict__ deg, const float* __restrict__ W,
    const float* __restrict__ bias, float* __restrict__ out, int n_dst)
{
    __shared__ float feat[TILE_M * LDF];

    const int t    = threadIdx.x;
    const int row0 = blockIdx.x * TILE_M;

    // Stage concatenated features: feat[r][0:128]=h_d, feat[r][128:256]=mean.
    // Clamp OOB rows to a valid row (stores are guarded later).
    for (int i = t; i < TILE_M * KDIM; i += 256) {
        const int r = i >> 8;           // / KDIM
        const int c = i & (KDIM - 1);
        int g = row0 + r;
        if (g >= n_dst) g = n_dst - 1;
        float v;
        if (c < IN_FEAT) {
            v = h_d[(size_t)g * IN_FEAT + c];
        } else {
            const float d = deg[g];
            v = nsum[(size_t)g * IN_FEAT + (c - IN_FEAT)] / fmaxf(d, 1.0f);
        }
        feat[r * LDF + c] = v;
    }
    __syncthreads();

    const int wave  = t >> 5;           // 0..7 -> output column tile
    const int lane  = t & 31;
    const int n     = wave * 16 + (lane & 15);  // global output column
    const int m     = lane & 15;                // A-matrix row within tile
    const int khalf = (lane >> 4) * 2;          // K sub-offset {0,2}

    const float* frow = feat + m * LDF;
    const float* wrow = W + (size_t)n * KDIM;   // W[n][*] == (W^T)[:, n]

    v8f acc = {};
#pragma unroll 4
    for (int k0 = 0; k0 < KDIM; k0 += 4) {
        v2f a = *(const v2f*)(frow + k0 + khalf);   // ds_load_b64
        v2f b = *(const v2f*)(wrow + k0 + khalf);   // global_load_b64 (L2 hit)
        acc = __builtin_amdgcn_wmma_f32_16x16x4_f32(
            /*neg_a=*/false, a, /*neg_b=*/false, b,
            /*c_mod=*/(short)0, acc, /*reuse_a=*/false, /*reuse_b=*/false);
    }

    const float bv    = bias[n];
    const int   mhigh = (lane >> 4) * 8;

    // Guarded stores only after all WMMAs (EXEC must be all-ones during WMMA).
#pragma unroll
    for (int v = 0; v < 8; ++v) {
        const int g = row0 + mhigh + v;
        if (g < n_dst) out[(size_t)g * OUT_FEAT + n] = acc[v] + bv;
    }
}

// ---------------------------------------------------------------------------
// Launcher
// ---------------------------------------------------------------------------
extern "C" void kernel_launch(void* const* d_in, const int* in_sizes, int n_in,
                              void* d_out, int out_size, void* d_ws, size_t ws_size,
                              hipStream_t stream) {
    const float* h_s  = (const float*)d_in[0];
    const float* h_d  = (const float*)d_in[1];
    const int*   src  = (const int*)d_in[2];
    const int*   dst  = (const int*)d_in[3];
    const float* W    = (const float*)d_in[4];
    const float* bias = (const float*)d_in[5];

    const int n_dst   = in_sizes[1] / IN_FEAT;
    const int n_edges = in_sizes[2];
    (void)n_in; (void)out_size; (void)ws_size;

    float* nsum = (float*)d_ws;                          // [n_dst, 128]
    float* deg  = nsum + (size_t)n_dst * IN_FEAT;        // [n_dst]

    const size_t zero_bytes = ((size_t)n_dst * IN_FEAT + (size_t)n_dst) * sizeof(float);
    hipMemsetAsync(d_ws, 0, zero_bytes, stream);

    sage_scatter<<<1024, 256, 0, stream>>>(h_s, src, dst, nsum, deg, n_edges);

    const int blocks = (n_dst + TILE_M - 1) / TILE_M;
    sage_gemm<<<blocks, 256, 0, stream>>>(h_d, nsum, deg, W, bias,
                                          (float*)d_out, n_dst);
}